// Encoder_22265110463340
// MI455X (gfx1250) — compile-verified
//
#include <hip/hip_runtime.h>
#include <stdint.h>

// ---------------- problem constants (from reference) ----------------
#define BSZ       1024
#define SEQ       200
#define DMODEL    512
#define KCH       4                  // seq rows per WMMA step (K of 16x16x4)
#define NCHUNK    (SEQ / KCH)        // 50
#define ROWPAD    16                 // pad LDS rows to stagger banks
#define ROWSTRIDE (DMODEL + ROWPAD)  // 528 floats

typedef __attribute__((ext_vector_type(2))) float v2f;
typedef __attribute__((ext_vector_type(4))) float v4f;
typedef __attribute__((ext_vector_type(8))) float v8f;
typedef int v4i __attribute__((vector_size(4 * sizeof(int))));

#define AS1 __attribute__((address_space(1)))
#define AS3 __attribute__((address_space(3)))

#if defined(__HIP_DEVICE_COMPILE__) && __has_builtin(__builtin_amdgcn_global_load_async_to_lds_b128)
#define USE_ASYNC 1
#else
#define USE_ASYNC 0
#endif

#if defined(__HIP_DEVICE_COMPILE__) && __has_builtin(__builtin_amdgcn_wmma_f32_16x16x4_f32)
#define USE_WMMA 1
#else
#define USE_WMMA 0
#endif

template <int N>
static __device__ __forceinline__ void wait_async_le() {
#if USE_ASYNC
#if __has_builtin(__builtin_amdgcn_s_wait_asynccnt)
  __builtin_amdgcn_s_wait_asynccnt(N);
#else
  asm volatile("s_wait_asynccnt %0" ::"n"(N) : "memory");
#endif
#endif
}

// Stage one chunk of KCH embedding rows (gather!) into an LDS buffer and
// fetch the KCH per-row scales (uniform -> scalarized by the compiler).
static __device__ __forceinline__ void stage_chunk(
    const int* __restrict__ xs, const float* __restrict__ W,
    const float* __restrict__ freqs, int b, int l0, int tid,
    float* dstbuf, float* sc) {
  int rowidx[KCH];
#pragma unroll
  for (int r = 0; r < KCH; ++r) {
    int idx = xs[b * SEQ + l0 + r];   // uniform across block -> s_load
    rowidx[r] = idx;
    sc[r] = freqs[idx];               // uniform -> s_load
  }
  // chunk = KCH*DMODEL floats = 512 x float4; 256 threads x 2 async B128 each
#pragma unroll
  for (int s = 0; s < 2; ++s) {
    int q = tid + s * 256;
    int r = q >> 7;                   // row within chunk
    int c = (q & 127) << 2;           // float column
    const float* gsrc = W + (size_t)rowidx[r] * DMODEL + c;
    float* ldst = dstbuf + r * ROWSTRIDE + c;
#if USE_ASYNC
    __builtin_amdgcn_global_load_async_to_lds_b128(
        (AS1 v4i*)(uintptr_t)gsrc,
        (AS3 v4i*)(uint32_t)(uintptr_t)ldst, 0, 0);
#else
    *(v4f*)ldst = *(const v4f*)gsrc;
#endif
  }
}

// ---------------- norm: deterministic two-stage sum of squares ----------------
__global__ void __launch_bounds__(256) enc_norm_partial(
    const int* __restrict__ xs, const float* __restrict__ freqs,
    float* __restrict__ ws) {
  __shared__ float part[8];
  int i = blockIdx.x * 256 + threadIdx.x;      // 800*256 == BSZ*SEQ
  float v = freqs[xs[i]];
  v *= v;
#pragma unroll
  for (int off = 16; off > 0; off >>= 1) v += __shfl_down(v, off, 32);
  int lane = threadIdx.x & 31, wid = threadIdx.x >> 5;
  if (lane == 0) part[wid] = v;
  __syncthreads();
  if (threadIdx.x == 0) {
    float t = 0.f;
#pragma unroll
    for (int w = 0; w < 8; ++w) t += part[w];
    ws[1 + blockIdx.x] = t;
  }
}

__global__ void __launch_bounds__(256) enc_norm_finish(float* __restrict__ ws) {
  __shared__ float part[8];
  float s = 0.f;
  for (int i = threadIdx.x; i < (BSZ * SEQ) / 256; i += 256) s += ws[1 + i];
#pragma unroll
  for (int off = 16; off > 0; off >>= 1) s += __shfl_down(s, off, 32);
  int lane = threadIdx.x & 31, wid = threadIdx.x >> 5;
  if (lane == 0) part[wid] = s;
  __syncthreads();
  if (threadIdx.x == 0) {
    float t = 0.f;
#pragma unroll
    for (int w = 0; w < 8; ++w) t += part[w];
    ws[0] = t;
  }
}

// ---------------- main pooling kernel: async gather + f32 WMMA ----------------
__global__ void __launch_bounds__(256) enc_pool(
    const int* __restrict__ xs, const float* __restrict__ W,
    const float* __restrict__ freqs, const float* __restrict__ ws,
    float* __restrict__ out) {
  __shared__ float smem[2 * KCH * ROWSTRIDE];
  const int b = blockIdx.x;
  const int tid = threadIdx.x;
  const int lane = tid & 31;
  const int wid = tid >> 5;    // wave 0..7 -> owns d range [wid*64, wid*64+64)
  const int m = lane & 15;     // A-row / output d sub-index
  const int half = lane >> 4;  // 0: K=0,1   1: K=2,3   (f32 16x4 A layout)

  const float inv_norm = rsqrtf(ws[0]);

#if USE_WMMA
  v8f acc[4] = {};             // 4 d-tiles of 16 per wave, f32 accum
#else
  float facc0 = 0.f, facc1 = 0.f;
#endif

  float sc_cur[KCH];
  float sc_nxt[KCH] = {0.f, 0.f, 0.f, 0.f};

  // prologue: stage chunk 0 into buffer 0
  stage_chunk(xs, W, freqs, b, 0, tid, &smem[0], sc_cur);

  for (int ch = 0; ch < NCHUNK; ++ch) {
    const int cur = ch & 1;
    if (ch + 1 < NCHUNK) {
      // issue next chunk's gathers into the other buffer, then wait for
      // the current chunk (each thread has <=2 newer asyncs outstanding)
      stage_chunk(xs, W, freqs, b, (ch + 1) * KCH, tid,
                  &smem[(cur ^ 1) * KCH * ROWSTRIDE], sc_nxt);
      wait_async_le<2>();
    } else {
      wait_async_le<0>();
    }
    __syncthreads();  // everyone's gathers for chunk `ch` visible in LDS

    const float* base = &smem[cur * KCH * ROWSTRIDE];
#if USE_WMMA
    {
      const int r0 = half * 2;
      v2f bb;                      // B[k,n] = s[k], broadcast over N
      bb.x = sc_cur[r0 + 0];
      bb.y = sc_cur[r0 + 1];
#pragma unroll
      for (int t = 0; t < 4; ++t) {
        const int d0 = wid * 64 + t * 16;
        v2f a;                     // A[m,k] = E[l0+k, d0+m] (transposed read)
        a.x = base[(r0 + 0) * ROWSTRIDE + d0 + m];
        a.y = base[(r0 + 1) * ROWSTRIDE + d0 + m];
        acc[t] = __builtin_amdgcn_wmma_f32_16x16x4_f32(
            false, a, false, bb, (short)0, acc[t], false, false);
      }
    }
#else
#pragma unroll
    for (int r = 0; r < KCH; ++r) {
      facc0 += sc_cur[r] * base[r * ROWSTRIDE + tid];
      facc1 += sc_cur[r] * base[r * ROWSTRIDE + tid + 256];
    }
#endif
    __syncthreads();  // chunk `ch` buffer may be overwritten next iteration
#pragma unroll
    for (int r = 0; r < KCH; ++r) sc_cur[r] = sc_nxt[r];
  }

#if USE_WMMA
  // C/D layout: lanes 0-15 vgpr j -> (M=j, N=lane); lanes 16-31 -> (M=8+j).
  // All N columns identical (B broadcast), so lanes with m==0 hold the tile.
  if (m == 0) {
    float* op = out + (size_t)b * DMODEL + wid * 64 + half * 8;
#pragma unroll
    for (int t = 0; t < 4; ++t)
#pragma unroll
      for (int j = 0; j < 8; ++j) op[t * 16 + j] = acc[t][j] * inv_norm;
  }
#else
  out[(size_t)b * DMODEL + tid] = facc0 * inv_norm;
  out[(size_t)b * DMODEL + tid + 256] = facc1 * inv_norm;
#endif
}

extern "C" void kernel_launch(void* const* d_in, const int* in_sizes, int n_in,
                              void* d_out, int out_size, void* d_ws,
                              size_t ws_size, hipStream_t stream) {
  (void)in_sizes; (void)n_in; (void)out_size; (void)ws_size;
  const int* xs = (const int*)d_in[0];        // [1024,200] token ids (int32)
  const float* W = (const float*)d_in[1];     // [100000,512]
  const float* freqs = (const float*)d_in[2]; // [100000]
  float* out = (float*)d_out;                 // [1024,512]
  float* ws = (float*)d_ws;                   // ws[0]=sumsq, ws[1..800]=partials

  enc_norm_partial<<<(BSZ * SEQ) / 256, 256, 0, stream>>>(xs, freqs, ws);
  enc_norm_finish<<<1, 256, 0, stream>>>(ws);
  enc_pool<<<BSZ, 256, 0, stream>>>(xs, W, freqs, ws, out);
}